// RandomChunkFilter_14250701488172
// MI455X (gfx1250) — compile-verified
//
#include <hip/hip_runtime.h>
#include <math.h>

typedef float v2f __attribute__((ext_vector_type(2)));
typedef float v4f __attribute__((ext_vector_type(4)));
typedef float v8f __attribute__((ext_vector_type(8)));

#define CHUNKC 128
#define SAMPLEC 16
#define BB 8
#define TT 8192
#define DD 1024
#define HH 1024
#define NCHUNKS (TT / CHUNKC)   /* 64  */
#define SMAX (SAMPLEC * CHUNKC) /* 2048 */
#define KB 32                   /* K-block staged in LDS */
#define LDA 34                  /* LDS row stride for A tile (floats) */
#define LDW2 288                /* LDS pair-row stride for W tile (floats): 128*2 + 32 pad */

// ---------------------------------------------------------------------------
// Kernel 1: per-batch unique-chunk selection (ascending order), count in ws.
// ws layout: sel[b*32 + 0] = n_sel ; sel[b*32 + 1 .. ] = sorted chunk ids
// ---------------------------------------------------------------------------
__global__ void rcf_select_kernel(const int* __restrict__ chunk_idx,
                                  int* __restrict__ sel) {
  const int b = blockIdx.x;
  if (threadIdx.x == 0) {
    unsigned long long m = 0ull;
    for (int i = 0; i < SAMPLEC; ++i) {
      m |= 1ull << (chunk_idx[b * SAMPLEC + i] & 63);
    }
    int n = 0;
    for (int c = 0; c < NCHUNKS; ++c) {
      if ((m >> c) & 1ull) sel[b * 32 + 1 + (n++)] = c;
    }
    sel[b * 32] = n;
  }
}

// ---------------------------------------------------------------------------
// Kernel 2: gathered GEMM + bias + sinusoidal PE + sqrt(512) scale.
// grid = (HH/128 n-tiles, SAMPLE slots, B batches), 256 threads = 8 waves.
// Each block: 128(M) x 128(N) tile, K=1024, register-double-buffered LDS
// staging, f32 WMMA 16x16x4. W tile K-pair interleaved in LDS so each B
// fragment is a single aligned ds_load_b64.
// ---------------------------------------------------------------------------
__launch_bounds__(256)
__global__ void rcf_gemm_kernel(const float* __restrict__ data,
                                const float* __restrict__ Wm,
                                const float* __restrict__ bias,
                                const int* __restrict__ sel,
                                float* __restrict__ out) {
  __shared__ float lA[CHUNKC * LDA];      // 128 x 32 tile of data (padded)
  __shared__ float lW[(KB / 2) * LDW2];   // 16 pair-rows x (128 cols x 2)

  const int b    = blockIdx.z;
  const int slot = blockIdx.y;
  const int n0   = blockIdx.x * 128;
  const int tid  = threadIdx.x;

  const int n_sel = sel[b * 32];
  float* outTile = out + ((size_t)b * SMAX + (size_t)slot * CHUNKC) * HH + n0;

  if (slot >= n_sel) {
    // Entire 128x128 tile is past `lens`: write zeros (valid-mask region).
    v4f z4 = {0.f, 0.f, 0.f, 0.f};
    for (int i = tid; i < (CHUNKC * 128) / 4; i += 256) {
      const int r = (i * 4) / 128;
      const int c = (i * 4) % 128;
      *(v4f*)(outTile + (size_t)r * HH + c) = z4;
    }
    return;
  }

  const int chunk = sel[b * 32 + 1 + slot];
  const int t0 = chunk * CHUNKC;
  const float* Abase = data + ((size_t)b * TT + t0) * DD;

  const int wave = tid >> 5;   // 8 waves (wave32)
  const int lane = tid & 31;
  const int hi   = (lane >> 4) & 1;
  const int ln   = lane & 15;
  const int m0   = wave * 16;  // wave's 16-row strip

  v8f acc[8];
  const v8f zero8 = {0.f, 0.f, 0.f, 0.f, 0.f, 0.f, 0.f, 0.f};
#pragma unroll
  for (int n = 0; n < 8; ++n) acc[n] = zero8;

  // cooperative load mappings
  const int arow = tid >> 1;        // A: 2 threads per 32-float row
  const int acol = (tid & 1) * 16;  //    each thread 16 floats (4x float4)
  const int pr   = tid >> 4;        // W: pair-row (rows 2pr, 2pr+1); 16 pairs
  const int wc   = (tid & 15) * 8;  //    8 columns per thread

  const float* aptr = Abase + (size_t)arow * DD + acol;
  const float* wptr = Wm + (size_t)(2 * pr) * HH + n0 + wc;

  // ---- prologue: fetch K-block 0 into registers ----
  v4f ra[4], rw[4];
#pragma unroll
  for (int i = 0; i < 4; ++i) ra[i] = *(const v4f*)(aptr + i * 4);
  rw[0] = *(const v4f*)(wptr);
  rw[1] = *(const v4f*)(wptr + 4);
  rw[2] = *(const v4f*)(wptr + HH);
  rw[3] = *(const v4f*)(wptr + HH + 4);

  for (int k0 = 0; k0 < DD; k0 += KB) {
    // ---- commit registers -> LDS ----
#pragma unroll
    for (int i = 0; i < 4; ++i) {
      *(v4f*)(&lA[arow * LDA + acol + i * 4]) = ra[i];
    }
    {
      float* wd = &lW[pr * LDW2 + wc * 2];
      v4f s0 = {rw[0].x, rw[2].x, rw[0].y, rw[2].y};
      v4f s1 = {rw[0].z, rw[2].z, rw[0].w, rw[2].w};
      v4f s2 = {rw[1].x, rw[3].x, rw[1].y, rw[3].y};
      v4f s3 = {rw[1].z, rw[3].z, rw[1].w, rw[3].w};
      *(v4f*)(wd + 0)  = s0;
      *(v4f*)(wd + 4)  = s1;
      *(v4f*)(wd + 8)  = s2;
      *(v4f*)(wd + 12) = s3;
    }
    __syncthreads();

    // ---- issue next K-block's global loads (overlap with compute) ----
    if (k0 + KB < DD) {
      const float* an = aptr + (k0 + KB);
      const float* wn = wptr + (size_t)(k0 + KB) * HH;
#pragma unroll
      for (int i = 0; i < 4; ++i) ra[i] = *(const v4f*)(an + i * 4);
      rw[0] = *(const v4f*)(wn);
      rw[1] = *(const v4f*)(wn + 4);
      rw[2] = *(const v4f*)(wn + HH);
      rw[3] = *(const v4f*)(wn + HH + 4);
    }

    // ---- compute: 8 kk-steps x 8 n-tiles of f32 WMMA ----
#pragma unroll
    for (int kk = 0; kk < KB; kk += 4) {
      // A fragment (16x4 f32): lanes 0-15 K=kk,kk+1 ; lanes 16-31 K=kk+2,kk+3
      v2f a = *(const v2f*)(&lA[(m0 + ln) * LDA + kk + 2 * hi]);
      const float* wb = &lW[((kk >> 1) + hi) * LDW2 + ln * 2];
#pragma unroll
      for (int n = 0; n < 8; ++n) {
        // B fragment: one aligned b64 = {W[kk+2hi][h], W[kk+2hi+1][h]}
        v2f bf = *(const v2f*)(wb + n * 32);
        acc[n] = __builtin_amdgcn_wmma_f32_16x16x4_f32(
            false, a, false, bf, (short)0, acc[n], false, false);
      }
    }
    __syncthreads();
  }

  // Epilogue: (acc + bias + PE(t,h)) * sqrt(512)
  const float scale = 22.62741699796952f;             // sqrt(512)
  const float c0 = -9.210340371976184f / (float)HH;   // -ln(10000)/H
#pragma unroll
  for (int n = 0; n < 8; ++n) {
    const int h = n0 + n * 16 + ln;
    const float bv = bias[h];
    const float div = __expf((float)((h >> 1) * 2) * c0);
    const bool is_cos = (h & 1);
#pragma unroll
    for (int j = 0; j < 8; ++j) {
      const int m = j + 8 * hi;                        // C-matrix row layout
      const int t = t0 + m0 + m;
      const float ang = (float)t * div;
      const float pe = is_cos ? __cosf(ang) : __sinf(ang);
      outTile[(size_t)(m0 + m) * HH + h] = (acc[n][j] + bv + pe) * scale;
    }
  }
}

extern "C" void kernel_launch(void* const* d_in, const int* in_sizes, int n_in,
                              void* d_out, int out_size, void* d_ws, size_t ws_size,
                              hipStream_t stream) {
  // setup_inputs order: state, data, W, b, chunk_idx  (state is unused)
  const float* data = (const float*)d_in[1];
  const float* Wm   = (const float*)d_in[2];
  const float* bias = (const float*)d_in[3];
  const int* cidx   = (const int*)d_in[4];
  float* out = (float*)d_out;
  int* sel = (int*)d_ws; // B*32 ints

  rcf_select_kernel<<<BB, 32, 0, stream>>>(cidx, sel);

  dim3 grid(HH / 128, SAMPLEC, BB);
  rcf_gemm_kernel<<<grid, 256, 0, stream>>>(data, Wm, bias, sel, out);
}